// myGNN_hidden_16690242912991
// MI455X (gfx1250) — compile-verified
//
#include <hip/hip_runtime.h>
#include <hip/hip_bf16.h>
#include <math.h>

typedef __attribute__((ext_vector_type(2))) float v2f;
typedef __attribute__((ext_vector_type(8))) float v8f;

#define DD 64

// ---------------------------------------------------------------------------
// Scatter: agg[dst[e]] += edge_weight[e] * h[src[e]]   (one wave32 per edge,
// 2 floats per lane, hardware fp32 global atomics)
// ---------------------------------------------------------------------------
__global__ void gnn_scatter_kernel(const float* __restrict__ h,
                                   const int* __restrict__ src,
                                   const int* __restrict__ dst,
                                   const float* __restrict__ ew,
                                   float* __restrict__ agg,
                                   int n_edges) {
    int tid  = blockIdx.x * blockDim.x + threadIdx.x;
    int e    = tid >> 5;          // one wave (32 lanes) per edge
    int lane = tid & 31;
    if (e >= n_edges) return;

    int   s  = src[e];
    int   d  = dst[e];
    float wt = ew[e];

    const float2 hv = *(const float2*)(h + (size_t)s * DD + lane * 2);
    float* p = agg + (size_t)d * DD + lane * 2;
    unsafeAtomicAdd(p,     wt * hv.x);
    unsafeAtomicAdd(p + 1, wt * hv.y);
}

// ---------------------------------------------------------------------------
// GEMM: hout = agg @ Wrel + brel + hin @ Wroot   (+ tanh on last layer)
// One wave32 computes one 16x16 output tile via V_WMMA_F32_16X16X4_F32.
// 8 waves / 256-thread block; both 64x64 weight matrices staged in LDS.
// ---------------------------------------------------------------------------
__global__ void gnn_gemm_wmma_kernel(const float* __restrict__ agg,
                                     const float* __restrict__ hin,
                                     const float* __restrict__ Wrel,
                                     const float* __restrict__ brel,
                                     const float* __restrict__ Wroot,
                                     float* __restrict__ hout,
                                     int n_nodes, int n_tiles, int last) {
    __shared__ float lw[2][DD * DD];   // [0]=W_rel, [1]=W_root : 32 KB

    // cooperative LDS fill: 256 threads x 4 iters x float4 = 4096 floats/matrix
    for (int i = threadIdx.x * 4; i < DD * DD; i += 256 * 4) {
        *(float4*)&lw[0][i] = *(const float4*)&Wrel[i];
        *(float4*)&lw[1][i] = *(const float4*)&Wroot[i];
    }
    __syncthreads();

    const int wave = threadIdx.x >> 5;
    const int tile = blockIdx.x * 8 + wave;
    if (tile >= n_tiles) return;

    const int tm = tile >> 2;           // 4 column tiles (D=64 -> 4*16)
    const int tn = tile & 3;
    const int m0 = tm * 16;
    const int n0 = tn * 16;

    const int lane = threadIdx.x & 31;
    const int half = lane >> 4;         // 0: lanes 0-15, 1: lanes 16-31
    const int l16  = lane & 15;

    // A operand row (clamped so OOB tiles read valid memory; EXEC must stay all-1)
    int arow_i = m0 + l16;
    if (arow_i >= n_nodes) arow_i = n_nodes - 1;
    const float* arow = agg + (size_t)arow_i * DD;
    const float* hrow = hin + (size_t)arow_i * DD;

    const int col = n0 + l16;           // B column / C column for this lane

    v8f c = {};
    #pragma unroll
    for (int kb = 0; kb < DD; kb += 4) {
        const int ka = kb + half * 2;   // this lane's K pair

        // A = agg tile, B = W_rel
        const float2 a1v = *(const float2*)(arow + ka);
        v2f a1 = { a1v.x, a1v.y };
        v2f b1 = { lw[0][ka * DD + col], lw[0][(ka + 1) * DD + col] };
        c = __builtin_amdgcn_wmma_f32_16x16x4_f32(false, a1, false, b1,
                                                  (short)0, c, false, false);

        // A = h tile, B = W_root
        const float2 a2v = *(const float2*)(hrow + ka);
        v2f a2 = { a2v.x, a2v.y };
        v2f b2 = { lw[1][ka * DD + col], lw[1][(ka + 1) * DD + col] };
        c = __builtin_amdgcn_wmma_f32_16x16x4_f32(false, a2, false, b2,
                                                  (short)0, c, false, false);
    }

    const float bv = brel[col];
    #pragma unroll
    for (int v = 0; v < 8; ++v) {
        const int row = m0 + half * 8 + v;
        if (row >= n_nodes) continue;
        float r = c[v] + bv;
        if (last) r = tanhf(r);
        hout[(size_t)row * DD + col] = r;
    }
}

// ---------------------------------------------------------------------------
// Launch: 3 layers of { memset agg; scatter; wmma-gemm }, ping-ponging
// x -> hA -> hB -> d_out.
// ---------------------------------------------------------------------------
extern "C" void kernel_launch(void* const* d_in, const int* in_sizes, int n_in,
                              void* d_out, int out_size, void* d_ws, size_t ws_size,
                              hipStream_t stream) {
    const float* x     = (const float*)d_in[0];
    const int*   eidx  = (const int*)d_in[1];     // [2, E]
    const float* ew    = (const float*)d_in[2];   // [E]
    const float* Wrel  = (const float*)d_in[3];   // [L, D, D]
    const float* brel  = (const float*)d_in[4];   // [L, D]
    const float* Wroot = (const float*)d_in[5];   // [L, D, D]
    float*       out   = (float*)d_out;

    const int N = in_sizes[0] / DD;
    const int E = in_sizes[1] / 2;
    const int L = in_sizes[3] / (DD * DD);

    float* agg = (float*)d_ws;
    float* hA  = agg + (size_t)N * DD;
    float* hB  = hA  + (size_t)N * DD;

    const int n_tiles = ((N + 15) / 16) * (DD / 16);

    const dim3 sblk(256);
    const dim3 sgrd((unsigned)(((size_t)E * 32 + 255) / 256));
    const dim3 gblk(256);
    const dim3 ggrd((unsigned)((n_tiles + 7) / 8));

    const int*   src = eidx;
    const int*   dstv = eidx + E;

    const float* hcur = x;
    float* interm[2] = { hA, hB };

    for (int l = 0; l < L; ++l) {
        hipMemsetAsync(agg, 0, (size_t)N * DD * sizeof(float), stream);

        gnn_scatter_kernel<<<sgrd, sblk, 0, stream>>>(hcur, src, dstv, ew, agg, E);

        const int last = (l == L - 1);
        float* hout = last ? out : interm[l & 1];

        gnn_gemm_wmma_kernel<<<ggrd, gblk, 0, stream>>>(
            agg, hcur,
            Wrel + (size_t)l * DD * DD,
            brel + (size_t)l * DD,
            Wroot + (size_t)l * DD * DD,
            hout, N, n_tiles, last);

        hcur = hout;
    }
}